// Net_29832842838041
// MI455X (gfx1250) — compile-verified
//
#include <hip/hip_runtime.h>
#include <cstdint>

typedef __attribute__((ext_vector_type(2))) float v2f;
typedef __attribute__((ext_vector_type(8))) float v8f;

// ---------------------------------------------------------------------------
// Degree / normalization
// ---------------------------------------------------------------------------
__global__ void deg_init_kernel(float* __restrict__ deg, int N) {
    int i = blockIdx.x * blockDim.x + threadIdx.x;
    if (i < N) deg[i] = 1.0f;   // self-loop contributes 1 to every node
}

__global__ void deg_edges_kernel(const long long* __restrict__ dst,
                                 float* __restrict__ deg, long long E) {
    long long e = (long long)blockIdx.x * blockDim.x + threadIdx.x;
    if (e < E) atomicAdd(&deg[dst[e]], 1.0f);
}

__global__ void dinv_kernel(const float* __restrict__ deg,
                            float* __restrict__ dinv, int N) {
    int i = blockIdx.x * blockDim.x + threadIdx.x;
    if (i < N) {
        float d = deg[i];
        dinv[i] = (d > 0.0f) ? rsqrtf(d) : 0.0f;   // deg >= 1 always, kept for fidelity
    }
}

// ---------------------------------------------------------------------------
// GEMM1: h0[N,16] = x[N,512] @ W1[512,16]  -- exact fp32 via V_WMMA_F32_16X16X4_F32
// One wave computes one 16x16 tile; K looped 512/4 = 128 WMMA ops.
// f32 A 16x4 layout: lanes 0-15: {K=0,K=1} in vgpr{0,1}; lanes 16-31: {K=2,K=3}.
// f32 B 4x16 layout mirrors A with N on lanes.  C/D 16x16: vgpr r -> row r+8*(lane/16).
// ---------------------------------------------------------------------------
__global__ void gemm_x_w1_kernel(const float* __restrict__ x,
                                 const float* __restrict__ W1,
                                 float* __restrict__ h0, int N) {
    int wave = (int)((blockIdx.x * (long long)blockDim.x + threadIdx.x) >> 5);
    int lane = threadIdx.x & 31;
    int m0 = wave * 16;
    if (m0 >= N) return;                     // uniform per wave: EXEC stays all-1s
    int half = lane >> 4;                    // 0: lanes 0-15, 1: lanes 16-31
    int l16  = lane & 15;
    const float* arow = x + (size_t)(m0 + l16) * 512;

    v8f c = {0.f, 0.f, 0.f, 0.f, 0.f, 0.f, 0.f, 0.f};
    for (int k = 0; k < 512; k += 4) {
        int ka = k + half * 2;
        v2f a, b;
        a.x = arow[ka];
        a.y = arow[ka + 1];
        b.x = W1[(size_t)ka * 16 + l16];
        b.y = W1[(size_t)(ka + 1) * 16 + l16];
        c = __builtin_amdgcn_wmma_f32_16x16x4_f32(false, a, false, b,
                                                  (short)0, c, false, false);
    }
    float* out = h0 + (size_t)m0 * 16;
#pragma unroll
    for (int r = 0; r < 8; ++r)
        out[(size_t)(r + 8 * half) * 16 + l16] = c[r];
}

// ---------------------------------------------------------------------------
// Self-loop term doubles as full-buffer init: hout = hin * dinv[node]^2
// ---------------------------------------------------------------------------
template <int F>
__global__ void self_init_kernel(const float* __restrict__ hin,
                                 const float* __restrict__ dinv,
                                 float* __restrict__ hout, long long total) {
    long long i = (long long)blockIdx.x * blockDim.x + threadIdx.x;
    if (i < total) {
        float w = dinv[i / F];
        hout[i] = hin[i] * (w * w);
    }
}

// ---------------------------------------------------------------------------
// Edge aggregation: hout[dst] += hin[src] * dinv[src]*dinv[dst]
// One thread per (edge, 4-feature chunk); float4 gather, 4x global_atomic_add_f32
// ---------------------------------------------------------------------------
template <int F>
__global__ void agg_edges_kernel(const long long* __restrict__ src,
                                 const long long* __restrict__ dst,
                                 const float* __restrict__ dinv,
                                 const float* __restrict__ hin,
                                 float* __restrict__ hout, long long E) {
    constexpr int CH = F / 4;
    long long gid = (long long)blockIdx.x * blockDim.x + threadIdx.x;
    long long e = gid / CH;
    int c = (int)(gid % CH);
    if (e >= E) return;
    long long s = src[e];
    long long d = dst[e];
    float w = dinv[s] * dinv[d];
    const float* pin = hin + s * F + c * 4;
    __builtin_prefetch(pin, 0, 0);           // global_prefetch_b8
    float4 v = *(const float4*)pin;
    float* o = hout + d * F + c * 4;
    atomicAdd(o + 0, v.x * w);
    atomicAdd(o + 1, v.y * w);
    atomicAdd(o + 2, v.z * w);
    atomicAdd(o + 3, v.w * w);
}

// ---------------------------------------------------------------------------
// GEMM2: t[N,64] = relu(h1 + b1)[N,16] @ W2[16,64]; bias+ReLU fused into A load.
// Wave computes a 16x16 tile; 4 column-tiles cover C=64; K=16 -> 4 WMMA steps.
// ---------------------------------------------------------------------------
__global__ void gemm_h_w2_kernel(const float* __restrict__ h1,
                                 const float* __restrict__ W2,
                                 const float* __restrict__ b1,
                                 float* __restrict__ t, int N) {
    int wave = (int)((blockIdx.x * (long long)blockDim.x + threadIdx.x) >> 5);
    int lane = threadIdx.x & 31;
    int mt = wave >> 2;                      // row tile
    int nt = wave & 3;                       // column tile (0..3)
    int m0 = mt * 16;
    if (m0 >= N) return;                     // uniform per wave
    int n0 = nt * 16;
    int half = lane >> 4;
    int l16  = lane & 15;
    const float* arow = h1 + (size_t)(m0 + l16) * 16;

    v8f c = {0.f, 0.f, 0.f, 0.f, 0.f, 0.f, 0.f, 0.f};
#pragma unroll
    for (int k = 0; k < 16; k += 4) {
        int ka = k + half * 2;
        v2f a, b;
        a.x = fmaxf(arow[ka]     + b1[ka],     0.0f);
        a.y = fmaxf(arow[ka + 1] + b1[ka + 1], 0.0f);
        b.x = W2[(size_t)ka * 64 + n0 + l16];
        b.y = W2[(size_t)(ka + 1) * 64 + n0 + l16];
        c = __builtin_amdgcn_wmma_f32_16x16x4_f32(false, a, false, b,
                                                  (short)0, c, false, false);
    }
#pragma unroll
    for (int r = 0; r < 8; ++r)
        t[(size_t)(m0 + r + 8 * half) * 64 + n0 + l16] = c[r];
}

// ---------------------------------------------------------------------------
// log_softmax over C=64, one wave32 per node, 2 columns per lane. b2 added here.
// ---------------------------------------------------------------------------
__global__ void log_softmax_kernel(float* __restrict__ out,
                                   const float* __restrict__ b2, int N) {
    int wave = (int)((blockIdx.x * (long long)blockDim.x + threadIdx.x) >> 5);
    int lane = threadIdx.x & 31;
    if (wave >= N) return;
    float* row = out + (size_t)wave * 64;
    float v0 = row[lane]      + b2[lane];
    float v1 = row[lane + 32] + b2[lane + 32];

    float m = fmaxf(v0, v1);
#pragma unroll
    for (int off = 16; off > 0; off >>= 1)
        m = fmaxf(m, __shfl_xor(m, off, 32));

    float s = __expf(v0 - m) + __expf(v1 - m);
#pragma unroll
    for (int off = 16; off > 0; off >>= 1)
        s += __shfl_xor(s, off, 32);

    float lse = m + __logf(s);
    row[lane]      = v0 - lse;
    row[lane + 32] = v1 - lse;
}

// ---------------------------------------------------------------------------
// Host-side launcher
// ---------------------------------------------------------------------------
extern "C" void kernel_launch(void* const* d_in, const int* in_sizes, int n_in,
                              void* d_out, int out_size, void* d_ws, size_t ws_size,
                              hipStream_t stream) {
    const float*      x   = (const float*)d_in[0];
    const long long*  ei  = (const long long*)d_in[1];   // int64 per reference
    const float*      W1  = (const float*)d_in[2];
    const float*      b1  = (const float*)d_in[3];
    const float*      W2  = (const float*)d_in[4];
    const float*      b2  = (const float*)d_in[5];
    float*            out = (float*)d_out;

    const int       N = in_sizes[0] / 512;
    const long long E = (long long)in_sizes[1] / 2;
    const long long* src = ei;       // edge_index[0]
    const long long* dst = ei + E;   // edge_index[1]

    // Workspace layout (floats): deg[N] | dinv[N] | h0[16N] | h1[16N] | t[64N]
    float* ws   = (float*)d_ws;
    float* deg  = ws;
    float* dinv = deg  + N;
    float* h0   = dinv + N;
    float* h1   = h0   + (size_t)16 * N;
    float* t    = h1   + (size_t)16 * N;

    const int B = 256;
    auto blocks = [B](long long n) { return (int)((n + B - 1) / B); };

    // --- normalization ---
    deg_init_kernel<<<blocks(N), B, 0, stream>>>(deg, N);
    deg_edges_kernel<<<blocks(E), B, 0, stream>>>(dst, deg, E);
    dinv_kernel<<<blocks(N), B, 0, stream>>>(deg, dinv, N);

    // --- layer 1: dense transform + aggregate + (bias/relu fused later) ---
    long long waves1 = ((long long)N + 15) / 16;
    gemm_x_w1_kernel<<<blocks(waves1 * 32), B, 0, stream>>>(x, W1, h0, N);
    self_init_kernel<16><<<blocks((long long)N * 16), B, 0, stream>>>(h0, dinv, h1, (long long)N * 16);
    agg_edges_kernel<16><<<blocks(E * 4), B, 0, stream>>>(src, dst, dinv, h0, h1, E);

    // --- layer 2: dense transform (bias+relu fused into A load) + aggregate ---
    long long waves2 = (((long long)N + 15) / 16) * 4;
    gemm_h_w2_kernel<<<blocks(waves2 * 32), B, 0, stream>>>(h1, W2, b1, t, N);
    self_init_kernel<64><<<blocks((long long)N * 64), B, 0, stream>>>(t, dinv, out, (long long)N * 64);
    agg_edges_kernel<64><<<blocks(E * 16), B, 0, stream>>>(src, dst, dinv, t, out, E);

    // --- log_softmax (adds b2) ---
    log_softmax_kernel<<<blocks((long long)N * 32), B, 0, stream>>>(out, b2, N);
}